// EdgeLoss_89421219103312
// MI455X (gfx1250) — compile-verified
//
#include <hip/hip_runtime.h>

#define SQ2F 1.41421356237309504880f

typedef __attribute__((ext_vector_type(2))) float v2f;
typedef __attribute__((ext_vector_type(8))) float v8f;
typedef __attribute__((ext_vector_type(4))) int v4i;
typedef __attribute__((address_space(1))) v4i* gptr_v4i;
typedef __attribute__((address_space(3))) v4i* lptr_v4i;

// MAKE_NORMAL stored as MN[c][k] (4 x 16, k>=6 zero-padded), MN[c][k] = list[k][c]
__constant__ float MNc[4][16] = {
  { 0.f, -1.f, -SQ2F,  SQ2F,  SQ2F,  SQ2F, 0,0,0,0,0,0,0,0,0,0},
  { 1.f,  0.f, -SQ2F, -SQ2F, -SQ2F,  SQ2F, 0,0,0,0,0,0,0,0,0,0},
  { 0.f,  1.f,  SQ2F, -SQ2F, -SQ2F, -SQ2F, 0,0,0,0,0,0,0,0,0,0},
  {-1.f,  0.f,  SQ2F,  SQ2F,  SQ2F, -SQ2F, 0,0,0,0,0,0,0,0,0,0},
};

#if defined(__gfx1250__) && __has_builtin(__builtin_amdgcn_global_load_async_to_lds_b128) && __has_builtin(__builtin_amdgcn_s_wait_asynccnt)
#define USE_ASYNC 1
#else
#define USE_ASYNC 0
#endif

// ---- compile-time probes: prove the CDNA5 paths are actually emitted ------
#if defined(__gfx1250__) && !USE_ASYNC
#error "gfx1250: global_load_async_to_lds_b128 / s_wait_asynccnt builtins unavailable"
#endif
#if defined(__gfx1250__) && !__has_builtin(__builtin_amdgcn_tanhf)
#error "gfx1250: __builtin_amdgcn_tanhf (v_tanh_f32) unavailable"
#endif
#if defined(__gfx1250__) && !__has_builtin(__builtin_amdgcn_wmma_f32_16x16x4_f32)
#error "gfx1250: __builtin_amdgcn_wmma_f32_16x16x4_f32 unavailable"
#endif
// ---------------------------------------------------------------------------

__device__ __forceinline__ float sigmoidf_fast(float z) {
#if defined(__gfx1250__) && __has_builtin(__builtin_amdgcn_tanhf)
  // sigmoid(z) = 0.5 + 0.5*tanh(z/2): single V_TANH_F32 instead of exp+rcp
  return fmaf(__builtin_amdgcn_tanhf(z * 0.5f), 0.5f, 0.5f);
#else
  return 1.0f / (1.0f + __expf(-z));
#endif
}

// ---------------------------------------------------------------------------
// Setup: normals = edges @ MAKE_NORMAL via V_WMMA_F32_16X16X4_F32 (2 tiles),
// then per-b constants (scrambled reshape mapping) into ws[b*16 .. b*16+15]:
// [ny0,nx0,ny1,nx1,ny2,nx2, dl0,dl1,dl2, dr0,dr1,dr2, 1/sizes[b][0], pad..]
// ---------------------------------------------------------------------------
__global__ void edge_setup_kernel(const float* __restrict__ edges,
                                  float* __restrict__ ws) {
  __shared__ float E[32][4];
  __shared__ float nm[32][16];
  __shared__ float n3n[32][6];
  __shared__ float dln[32][3], drn[32][3], invs[32];

  const int lane = threadIdx.x;  // 0..31, one full wave32, EXEC all ones
#pragma unroll
  for (int c = 0; c < 4; c++) E[lane][c] = edges[lane * 4 + c];
  __syncthreads();

#if defined(__gfx1250__) && __has_builtin(__builtin_amdgcn_wmma_f32_16x16x4_f32)
  {
    const int half = lane >> 4;   // 0: K=0,1  1: K=2,3
    const int l16  = lane & 15;
    v2f bop;
    bop.x = MNc[half * 2 + 0][l16];
    bop.y = MNc[half * 2 + 1][l16];
#pragma unroll
    for (int t = 0; t < 2; t++) {        // rows 0..15 and 16..31 of edges
      v2f aop;
      aop.x = E[t * 16 + l16][half * 2 + 0];
      aop.y = E[t * 16 + l16][half * 2 + 1];
      v8f c = {};
      c = __builtin_amdgcn_wmma_f32_16x16x4_f32(false, aop, false, bop,
                                                (short)0, c, false, false);
#pragma unroll
      for (int i = 0; i < 8; i++)        // D[m][n]: m = i + 8*half, n = l16
        nm[t * 16 + i + 8 * half][l16] = c[i];
    }
  }
#else
  for (int k = 0; k < 6; k++) {
    float s = 0.f;
    for (int c = 0; c < 4; c++) s += E[lane][c] * MNc[c][k];
    nm[lane][k] = s;
  }
#endif
  __syncthreads();

  {  // per-n: normalize, dists, inverse size
    const int n = lane;
    float N6[6], sz0 = 1.f;
#pragma unroll
    for (int e = 0; e < 3; e++) {
      float a = nm[n][2 * e], b = nm[n][2 * e + 1];
      float s = a * a + b * b;
      if (e == 0) sz0 = s;
      float r = 1.0f / sqrtf(s);
      N6[2 * e] = a * r;
      N6[2 * e + 1] = b * r;
    }
#pragma unroll
    for (int k = 0; k < 6; k++) n3n[n][k] = N6[k];
    // IDX_L = [0,1,0,1,2,3], IDX_R = [0,1,2,3,0,1]
    dln[n][0] = E[n][0] * N6[0] + E[n][1] * N6[1];
    dln[n][1] = E[n][0] * N6[2] + E[n][1] * N6[3];
    dln[n][2] = E[n][2] * N6[4] + E[n][3] * N6[5];
    drn[n][0] = E[n][0] * N6[0] + E[n][1] * N6[1];
    drn[n][1] = E[n][2] * N6[2] + E[n][3] * N6[3];
    drn[n][2] = E[n][0] * N6[4] + E[n][1] * N6[5];
    invs[n] = 1.0f / sz0;
  }
  __syncthreads();

  {  // gather the reshape-scrambled constants for batch slice b = lane
    const int b = lane;
    float out[16];
#pragma unroll
    for (int e2 = 0; e2 < 3; e2++) {
      int idx = e2 * 32 + b;
      int n = idx / 3, e = idx % 3;
      out[2 * e2]     = n3n[n][2 * e];      // ny
      out[2 * e2 + 1] = n3n[n][2 * e + 1];  // nx
      out[6 + e2]     = dln[n][e];
      out[9 + e2]     = drn[n][e];
    }
    out[12] = invs[b];
    out[13] = out[14] = out[15] = 0.f;
#pragma unroll
    for (int i = 0; i < 16; i++) ws[b * 16 + i] = out[i];
  }
}

// ---------------------------------------------------------------------------
// Main: stream target (b128) through async-LDS double buffer, per-pixel SDF +
// tanh-sigmoid, block tree-reduce -> one partial per block.
// Grid: 2048 blocks = 32 b-slices x 64 chunks; chunk = 1024 float4 (16 KB).
// ---------------------------------------------------------------------------
__global__ __launch_bounds__(256) void edge_loss_main(
    const float* __restrict__ target, const float* __restrict__ cst_ws,
    float* __restrict__ partials) {
  const int tid   = threadIdx.x;
  const int b     = blockIdx.x >> 6;
  const int chunk = blockIdx.x & 63;

  const float* cst = cst_ws + b * 16;
  const float ny0 = cst[0], nx0 = cst[1], ny1 = cst[2], nx1 = cst[3];
  const float ny2 = cst[4], nx2 = cst[5];
  const float dl0 = cst[6], dl1 = cst[7], dl2 = cst[8];
  const float dr0 = cst[9], dr1 = cst[10], dr2 = cst[11];
  const float invsz = cst[12];

  const float4* __restrict__ tgt4 =
      (const float4*)target + (size_t)b * 65536u + (size_t)chunk * 1024u;

  __shared__ float4 buf[2][256];
  __shared__ float red[256];
  float acc = 0.0f;

  auto compute = [&](float4 tv, int q4) {
    // pixel p = q4*4 within this b-slice: y = p>>9, x = p&511 (4 share y)
    const float fy = (float)(q4 >> 7);
    float a0 = fmaf(fy, ny0, (float)((q4 & 127) << 2) * nx0);
    float a1 = fmaf(fy, ny1, (float)((q4 & 127) << 2) * nx1);
    float a2 = fmaf(fy, ny2, (float)((q4 & 127) << 2) * nx2);
    const float* tp = (const float*)&tv;
#pragma unroll
    for (int j = 0; j < 4; j++) {
      float sl = fminf(fminf(a0 - dl0, a1 - dl1), a2 - dl2);
      float sr = fminf(fminf(dr0 - a0, dr1 - a1), dr2 - a2);
      float t = tp[j];
      acc += sigmoidf_fast(sl) * t + sigmoidf_fast(sr) * (1.0f - t);
      a0 += nx0; a1 += nx1; a2 += nx2;
    }
  };

#if USE_ASYNC
#define ASYNC_ISSUE(k)                                            \
  __builtin_amdgcn_global_load_async_to_lds_b128(                 \
      (gptr_v4i)(tgt4 + (k)*256 + tid),                           \
      (lptr_v4i)&buf[(k) & 1][tid], 0, 0)
  ASYNC_ISSUE(0);
#pragma unroll
  for (int i = 0; i < 4; i++) {
    if (i < 3) {
      ASYNC_ISSUE(i + 1);
      __builtin_amdgcn_s_wait_asynccnt(1);  // current tile landed in LDS
    } else {
      __builtin_amdgcn_s_wait_asynccnt(0);
    }
    __asm__ __volatile__("" ::: "memory");
    float4 tv = buf[i & 1][tid];            // wave-local: own lane's 16B only
    compute(tv, chunk * 1024 + i * 256 + tid);
  }
#undef ASYNC_ISSUE
#else
#pragma unroll
  for (int i = 0; i < 4; i++) {
    float4 tv = tgt4[i * 256 + tid];
    compute(tv, chunk * 1024 + i * 256 + tid);
  }
#endif

  red[tid] = acc;
  __syncthreads();
#pragma unroll
  for (int s = 128; s > 0; s >>= 1) {
    if (tid < s) red[tid] += red[tid + s];
    __syncthreads();
  }
  if (tid == 0) partials[blockIdx.x] = red[0] * invsz;  // sign applied at end
}

// ---------------------------------------------------------------------------
// Final: deterministic reduction of 2048 partials -> -sum / (32*512*512)
// ---------------------------------------------------------------------------
__global__ __launch_bounds__(256) void edge_loss_final(
    const float* __restrict__ partials, float* __restrict__ out) {
  __shared__ double red[256];
  const int tid = threadIdx.x;
  double s = 0.0;
  for (int i = tid; i < 2048; i += 256) s += (double)partials[i];
  red[tid] = s;
  __syncthreads();
#pragma unroll
  for (int k = 128; k > 0; k >>= 1) {
    if (tid < k) red[tid] += red[tid + k];
    __syncthreads();
  }
  if (tid == 0) out[0] = (float)(-red[0] / 8388608.0);
}

extern "C" void kernel_launch(void* const* d_in, const int* in_sizes, int n_in,
                              void* d_out, int out_size, void* d_ws,
                              size_t ws_size, hipStream_t stream) {
  const float* edges  = (const float*)d_in[0];   // (32, 4) fp32
  const float* target = (const float*)d_in[1];   // (32, 512, 512) fp32
  float* ws = (float*)d_ws;
  float* cst = ws;              // 512 floats: 32 x 16 per-b constants
  float* partials = ws + 512;   // 2048 floats: per-block partial sums
  float* out = (float*)d_out;   // 1 float

  edge_setup_kernel<<<dim3(1), dim3(32), 0, stream>>>(edges, cst);
  edge_loss_main<<<dim3(2048), dim3(256), 0, stream>>>(target, cst, partials);
  edge_loss_final<<<dim3(1), dim3(256), 0, stream>>>(partials, out);
}